// HeadedRepeatCausalLinear_27230092656747
// MI455X (gfx1250) — compile-verified
//
#include <hip/hip_runtime.h>
#include <cstdint>

// Problem constants (from reference): HEADS = HEAD_DIM = 4096, DIM = 2048, B = 1.
#define NN   4096      // heads == head_dim
#define H2   2048
#define TILE 64
#define LSTR 68        // padded LDS row stride in floats: 68*4 = 272 B, 16B aligned

// ---------------------------------------------------------------------------
// Prep: compact the strided columns weight[:,idx], bias[:,idx] into d_ws and
// compute the clipped decay scalar. 16 blocks x 256 threads.
// ---------------------------------------------------------------------------
__global__ __launch_bounds__(256) void hrcl_prep(
    const float* __restrict__ weight, const float* __restrict__ bias,
    const float* __restrict__ decay, const int* __restrict__ index,
    float* __restrict__ wcol, float* __restrict__ bcol, float* __restrict__ dval)
{
    const int k = blockIdx.x * blockDim.x + threadIdx.x;
    const int idx = *index;
    if (k < NN) {
        wcol[k] = weight[(size_t)k * H2 + idx];
        bcol[k] = bias  [(size_t)k * H2 + idx];
    }
    if (k == 0) {
        // d = clip(decay[1], 0.9, 1.0) ** (1/DECAY_CONSTANT), DECAY_CONSTANT = 1
        dval[0] = fminf(fmaxf(decay[1], 0.9f), 1.0f);
    }
}

// ---------------------------------------------------------------------------
// Main: 64x64 tile transpose-fused elementwise kernel.
//   xs[r][c] = x[i0+r][k0+c]        (row-major in i)
//   cs[r][c] = cache[k0+r][i0+c]    (row-major in k)
// out[i,k]       needs xs direct + cs transposed
// new_cache[k,i] needs cs direct + xs transposed
// Tiles are DMA'd into LDS with CDNA5 async-to-LDS loads (ASYNCcnt).
// ---------------------------------------------------------------------------
__global__ __launch_bounds__(256) void hrcl_main(
    const float* __restrict__ x, const float* __restrict__ cache,
    const float* __restrict__ wcol, const float* __restrict__ bcol,
    const float* __restrict__ dval,
    float* __restrict__ out, float* __restrict__ newc)
{
    __shared__ float xs[TILE * LSTR];
    __shared__ float cs[TILE * LSTR];

    const int tid = threadIdx.x;
    const int cx  = (tid & 15) << 2;   // column within tile: 0..60 step 4
    const int ry  = tid >> 4;          // row phase: 0..15
    const int i0  = blockIdx.y * TILE;
    const int k0  = blockIdx.x * TILE;

    // LDS byte addresses: generic pointers to __shared__ carry the LDS offset
    // in their low 32 bits (aperture in the high bits) -> truncate.
    const uint32_t xs_base = (uint32_t)(uintptr_t)(&xs[0]);
    const uint32_t cs_base = (uint32_t)(uintptr_t)(&cs[0]);

    // Async DMA both tiles into LDS: 16B per lane per op, fully coalesced,
    // LDS destinations 16B aligned (LSTR*4 = 272 is a multiple of 16).
    #pragma unroll
    for (int rr = 0; rr < 4; ++rr) {
        const int r = ry + rr * 16;
        const uint64_t gx = (uint64_t)(x     + (size_t)(i0 + r) * NN + (k0 + cx));
        const uint64_t gc = (uint64_t)(cache + (size_t)(k0 + r) * NN + (i0 + cx));
        const uint32_t lx = xs_base + (uint32_t)((r * LSTR + cx) * 4);
        const uint32_t lc = cs_base + (uint32_t)((r * LSTR + cx) * 4);
        asm volatile("global_load_async_to_lds_b128 %0, %1, off"
                     :: "v"(lx), "v"(gx) : "memory");
        asm volatile("global_load_async_to_lds_b128 %0, %1, off"
                     :: "v"(lc), "v"(gc) : "memory");
    }
    asm volatile("s_wait_asynccnt 0x0" ::: "memory");
    __syncthreads();

    const float d  = *dval;
    const bool  lo = (k0 < H2);        // block-uniform: 64 | 2048

    // Column params for the out phase (k = k0 + cx + j)
    const float4 wk4 = *(const float4*)(wcol + k0 + cx);
    const float4 bk4 = *(const float4*)(bcol + k0 + cx);
    const float wa[4] = {wk4.x, wk4.y, wk4.z, wk4.w};
    const float ba[4] = {bk4.x, bk4.y, bk4.z, bk4.w};

    #pragma unroll
    for (int rr = 0; rr < 4; ++rr) {
        const int r = ry + rr * 16;

        // ---- out tile: row i = i0+r, cols k = k0+cx .. +3 (coalesced B128 store)
        {
            const float4 xv = *(const float4*)(&xs[r * LSTR + cx]);
            const float xa[4] = {xv.x, xv.y, xv.z, xv.w};
            float oa[4];
            #pragma unroll
            for (int j = 0; j < 4; ++j) {
                const float cv = cs[(cx + j) * LSTR + r];   // cache[k,i] transposed
                float t;
                if (lo) { t = xa[j] + d * cv;         oa[j] = wa[j] * t + ba[j]; }
                else    { t = wa[j] * xa[j] + d * cv; oa[j] = t + ba[j]; }
            }
            *(float4*)(out + (size_t)(i0 + r) * NN + (k0 + cx)) =
                make_float4(oa[0], oa[1], oa[2], oa[3]);
        }

        // ---- new_cache tile: row k = k0+r, cols i = i0+cx .. +3 (coalesced B128 store)
        {
            const float4 cv = *(const float4*)(&cs[r * LSTR + cx]);
            const float ca[4] = {cv.x, cv.y, cv.z, cv.w};
            const float wk = wcol[k0 + r];
            float na[4];
            #pragma unroll
            for (int j = 0; j < 4; ++j) {
                const float xv = xs[(cx + j) * LSTR + r];   // x[i,k] transposed
                na[j] = lo ? (xv + d * ca[j]) : (wk * xv + d * ca[j]);
            }
            *(float4*)(newc + (size_t)(k0 + r) * NN + (i0 + cx)) =
                make_float4(na[0], na[1], na[2], na[3]);
        }
    }
}

extern "C" void kernel_launch(void* const* d_in, const int* in_sizes, int n_in,
                              void* d_out, int out_size, void* d_ws, size_t ws_size,
                              hipStream_t stream) {
    const float* x      = (const float*)d_in[0];
    const float* weight = (const float*)d_in[1];
    const float* bias   = (const float*)d_in[2];
    const float* decay  = (const float*)d_in[3];
    const float* cache  = (const float*)d_in[4];
    const int*   index  = (const int*)d_in[5];

    float* out  = (float*)d_out;                      // (4096, 4096)
    float* newc = out + (size_t)NN * NN;              // (4096, 4096)

    float* wcol = (float*)d_ws;                       // 4096 floats
    float* bcol = wcol + NN;                          // 4096 floats
    float* dval = bcol + NN;                          // 1 float

    hrcl_prep<<<NN / 256, 256, 0, stream>>>(weight, bias, decay, index, wcol, bcol, dval);

    dim3 grid(NN / TILE, NN / TILE);                  // 64 x 64 tiles
    hrcl_main<<<grid, 256, 0, stream>>>(x, cache, wcol, bcol, dval, out, newc);
}